// Large_batch_queue_classwise_26972394619742
// MI455X (gfx1250) — compile-verified
//
#include <hip/hip_runtime.h>
#include <stdint.h>

// ---------------------------------------------------------------------------
// Large_batch_queue_classwise for MI455X (gfx1250, wave32)
//
// Pass 1: last[C*Q] = -1                                   (d_ws scratch)
// Pass 2: exact occurrence rank occ_i via LDS-tiled O(N^2/2) count,
//         8-way j-split across the 8 waves of a 256-thread block,
//         then atomicMax(last[l*Q + (tail[l]+occ)%Q], i)   (deterministic)
// Pass 3: stream every queue bucket row (1 KB) to d_out; winner rows gather
//         from features. Copy path = CDNA5 async global->LDS->global
//         (global_load_async_to_lds_b128 / global_store_async_from_lds_b128,
//          s_wait_asynccnt), 4 x 16 B per lane in flight between waits.
//         ~820 MB HBM traffic => ~35 us floor at 23.3 TB/s; occ pass is
//         VALU-bound at ~8K cmp-adds worst-lane (~10 us), below the floor.
// ---------------------------------------------------------------------------

#define TILE 256
#define GRP  4   // row-groups per block in the output pass (async MLP batch)

__global__ void init_last_kernel(int* __restrict__ last, int n) {
  int i = blockIdx.x * blockDim.x + threadIdx.x;
  if (i < n) last[i] = -1;
}

// Block = 256 threads = 8 waves; block b owns proposals i in [32b, 32b+32).
// Wave w scans j-subrange [w*32, w*32+32) of each 256-wide label tile in LDS.
// Lane (tid&31) corresponds to i = 32b + lane.  occ reduced across waves.
__global__ void occ_scatter_kernel(const int* __restrict__ lab,
                                   const int* __restrict__ tail,
                                   int* __restrict__ last,
                                   int N, int C, int Q) {
  __shared__ int sLab[TILE];
  __shared__ int sPart[TILE];
  const int tid  = threadIdx.x;
  const int w    = tid >> 5;          // wave id 0..7
  const int lane = tid & 31;
  const int iBase = blockIdx.x * 32;
  const int i     = iBase + lane;
  const int myLab = (i < N) ? lab[i] : (int)0x80000000;

  int cnt = 0;
  const int iEnd  = min(iBase + 32, N);
  const int tiles = (iEnd + TILE - 1) / TILE;   // j in [0, tiles*256) covers [0, i]
  for (int t = 0; t < tiles; ++t) {
    __syncthreads();
    const int j0  = t * TILE;
    const int idx = j0 + tid;
    sLab[tid] = (idx < N) ? lab[idx] : (int)0x80000000;  // sentinel never matches valid label
    __syncthreads();
    if (j0 + TILE <= iBase) {
      // Fast path: every j in this tile is < i for all lanes -> equality only.
#pragma unroll
      for (int jj = 0; jj < 32; ++jj)
        cnt += (sLab[(w << 5) + jj] == myLab) ? 1 : 0;
    } else {
      // Diagonal tile: need the j < i predicate.
      const int jb = j0 + (w << 5);
#pragma unroll
      for (int jj = 0; jj < 32; ++jj) {
        const int lj = sLab[(w << 5) + jj];   // wave-uniform LDS broadcast
        cnt += ((lj == myLab) && ((jb + jj) < i)) ? 1 : 0;
      }
    }
  }
  sPart[tid] = cnt;
  __syncthreads();

  if (tid < 32 && i < N) {           // thread tid is (w=0, lane=tid) -> myLab is for this i
    int occ = 0;
#pragma unroll
    for (int ww = 0; ww < 8; ++ww) occ += sPart[(ww << 5) + tid];
    const int l = myLab;
    if (l >= 0 && l < C) {
      int tmod = tail[l] % Q; if (tmod < 0) tmod += Q;
      const int slot = (tmod + (occ % Q)) % Q;
      atomicMax(&last[l * Q + slot], i);   // last (max-index) writer wins
    }
  }
}

// 256 threads; tpr = F/4 threads per row (16 B per lane), rpb = 256/tpr rows per
// row-group, GRP row-groups per block.  Issue GRP async loads into disjoint LDS
// slots, one s_wait_asynccnt, then GRP async stores: 64 B per lane in flight.
__global__ void write_out_kernel(const float* __restrict__ features,
                                 const float* __restrict__ queue,
                                 const int* __restrict__ last,
                                 float* __restrict__ out,
                                 long long CQ, int F, int tpr, int rpb) {
  __shared__ __align__(16) char buf[256 * 16 * GRP];
  const int tid = threadIdx.x;
  const int c   = tid % tpr;                 // 16-byte chunk within a row
  const long long base = (long long)blockIdx.x * (rpb * GRP) + tid / tpr;

  unsigned long long gsrc[GRP], gdst[GRP];
  unsigned ldsoff[GRP];
  bool act[GRP];

#pragma unroll
  for (int g = 0; g < GRP; ++g) {
    const long long row = base + (long long)g * rpb;
    act[g] = (row < CQ);
    if (act[g]) {
      const int win = last[row];
      const float* src = (win >= 0) ? (features + (size_t)win * (size_t)F)
                                    : (queue    + (size_t)row * (size_t)F);
      gsrc[g]   = (unsigned long long)(uintptr_t)(src + (size_t)c * 4);
      gdst[g]   = (unsigned long long)(uintptr_t)(out + (size_t)row * (size_t)F
                                                      + (size_t)c * 4);
      ldsoff[g] = (unsigned)(uintptr_t)(&buf[(g * 256 + tid) * 16]);
    }
  }

#pragma unroll
  for (int g = 0; g < GRP; ++g)
    if (act[g])
      asm volatile("global_load_async_to_lds_b128 %0, %1, off"
                   :: "v"(ldsoff[g]), "v"(gsrc[g]) : "memory");

  asm volatile("s_wait_asynccnt 0x0" ::: "memory");

#pragma unroll
  for (int g = 0; g < GRP; ++g)
    if (act[g])
      asm volatile("global_store_async_from_lds_b128 %0, %1, off"
                   :: "v"(gdst[g]), "v"(ldsoff[g]) : "memory");

  asm volatile("s_wait_asynccnt 0x0" ::: "memory");
}

extern "C" void kernel_launch(void* const* d_in, const int* in_sizes, int n_in,
                              void* d_out, int out_size, void* d_ws, size_t ws_size,
                              hipStream_t stream) {
  const float* features = (const float*)d_in[0];   // (N, F) f32
  const float* queue    = (const float*)d_in[1];   // (C, Q, F) f32
  const int*   labels   = (const int*)d_in[2];     // (N,) i32
  const int*   tail     = (const int*)d_in[3];     // (C,) i32

  const int N = in_sizes[2];
  const int C = in_sizes[3];
  const int F = in_sizes[0] / N;
  const int Q = in_sizes[1] / (C * F);
  const long long CQ = (long long)C * Q;

  int* last = (int*)d_ws;                          // C*Q ints (1.6 MB) scratch

  // Pass 1: reset winner table (deterministic each call)
  {
    const int threads = 256;
    const int blocks  = (int)((CQ + threads - 1) / threads);
    init_last_kernel<<<blocks, threads, 0, stream>>>(last, (int)CQ);
  }
  // Pass 2: exact occurrence ranks + winner scatter
  {
    const int blocks = (N + 31) / 32;
    occ_scatter_kernel<<<blocks, 256, 0, stream>>>(labels, tail, last, N, C, Q);
  }
  // Pass 3: streamed output (every bucket row written)
  {
    const int tpr = F / 4;            // threads per row, 16 B each (F=256 -> 64)
    const int rpb = 256 / tpr;        // rows per row-group (-> 4)
    const long long rowsPerBlock = (long long)rpb * GRP;  // -> 16
    const int blocks = (int)((CQ + rowsPerBlock - 1) / rowsPerBlock);  // -> 25000
    write_out_kernel<<<blocks, 256, 0, stream>>>(features, queue, last,
                                                 (float*)d_out, CQ, F, tpr, rpb);
  }
}